// model_chain_14654428414669
// MI455X (gfx1250) — compile-verified
//
#include <hip/hip_runtime.h>
#include <hip/hip_bf16.h>

typedef float v2f __attribute__((ext_vector_type(2)));
typedef float v8f __attribute__((ext_vector_type(8)));

#define NEIG   16
#define NANG   254
#define BATCH  256
#define REFI   100
#define NPTS   256          // 100 bwd + 156 fwd curve points == K for splat GEMM
#define NPX    192
#define IMG    (NPX*NPX)    // 36864
#define INROW  (NEIG+7)     // 23

static __device__ __forceinline__ v8f wmma4(v2f a, v2f b, v8f c) {
  // V_WMMA_F32_16X16X4_F32 : full-precision f32 matrix core op (wave32)
  return __builtin_amdgcn_wmma_f32_16x16x4_f32(false, a, false, b, (short)0, c, false, false);
}

// ---------------------------------------------------------------- A1: psi/theta
__global__ void k_angles(const float* __restrict__ inp,
                         const float* __restrict__ Gpsi,
                         const float* __restrict__ Gth,
                         const float* __restrict__ Psi,
                         const float* __restrict__ Theta,
                         const int*   __restrict__ pidx,
                         float* __restrict__ psiB, float* __restrict__ thB) {
  int n = blockIdx.x * blockDim.x + threadIdx.x;
  if (n >= BATCH * NANG) return;
  int b = n / NANG, a = n % NANG;
  float dp = 0.f, dt = 0.f;
  #pragma unroll
  for (int e = 0; e < NEIG; ++e) {
    float v = inp[b * INROW + e];
    dp += v * Gpsi[a * NEIG + e];
    dt += v * Gth [a * NEIG + e];
  }
  int col = pidx[a];
  psiB[b * NANG + col] = Psi[col]   + dp;
  thB [b * NANG + col] = Theta[col] + dt;
}

// ---------------------------------------------------------------- A2: R = Rz(psi)Ry(theta)
__global__ void k_rots(const float* __restrict__ psiB, const float* __restrict__ thB,
                       float* __restrict__ R) {
  int n = blockIdx.x * blockDim.x + threadIdx.x;
  if (n >= BATCH * NANG) return;
  float p = psiB[n], t = thB[n];
  float c1 = cosf(p), s1 = sinf(p), c2 = cosf(t), s2 = sinf(t);
  float* r = R + (size_t)n * 9;
  r[0] =  c1 * c2; r[1] = -s1; r[2] = c1 * s2;
  r[3] =  s1 * c2; r[4] =  c1; r[5] = s1 * s2;
  r[6] = -s2;      r[7] = 0.f; r[8] = c2;
}

// ---------------------------------------------------------------- B: chain scans -> curve
static __device__ __forceinline__ void mm3(const float* A, const float* B, float* C) {
  #pragma unroll
  for (int i = 0; i < 3; ++i)
    #pragma unroll
    for (int j = 0; j < 3; ++j)
      C[i*3+j] = A[i*3+0]*B[0*3+j] + A[i*3+1]*B[1*3+j] + A[i*3+2]*B[2*3+j];
}
static __device__ __forceinline__ void mm3T(const float* A, const float* B, float* C) {
  // C = A^T * B
  #pragma unroll
  for (int i = 0; i < 3; ++i)
    #pragma unroll
    for (int j = 0; j < 3; ++j)
      C[i*3+j] = A[0*3+i]*B[0*3+j] + A[1*3+i]*B[1*3+j] + A[2*3+i]*B[2*3+j];
}

__global__ void k_scan(const float* __restrict__ inp,
                       const float* __restrict__ R,
                       const float* __restrict__ dists,
                       float* __restrict__ curve) {
  int b = blockIdx.x * blockDim.x + threadIdx.x;
  if (b >= BATCH) return;
  const float* row = inp + (size_t)b * INROW;
  float wx = row[NEIG+0], wy = row[NEIG+1], wz = row[NEIG+2];
  float x0 = row[NEIG+3], y0 = row[NEIG+4], z0 = row[NEIG+5];
  // Rodrigues
  float th = sqrtf(wx*wx + wy*wy + wz*wz) + 1e-8f;
  float kx = wx/th, ky = wy/th, kz = wz/th;
  float st = sinf(th), ct = 1.f - cosf(th);
  float F0[9];
  F0[0] = 1.f + ct*(-(ky*ky+kz*kz)); F0[1] = st*(-kz) + ct*(kx*ky);      F0[2] = st*(ky) + ct*(kx*kz);
  F0[3] = st*(kz) + ct*(kx*ky);      F0[4] = 1.f + ct*(-(kx*kx+kz*kz));  F0[5] = st*(-kx) + ct*(ky*kz);
  F0[6] = st*(-ky) + ct*(kx*kz);     F0[7] = st*(kx) + ct*(ky*kz);       F0[8] = 1.f + ct*(-(kx*kx+ky*ky));

  const float* Rb = R + (size_t)b * NANG * 9;
  float F[9], T[9];
  #pragma unroll
  for (int i = 0; i < 9; ++i) F[i] = F0[i];

  // forward: indices 100..255
  float gx = x0, gy = y0, gz = z0;
  float* cv = curve + (size_t)b * NPTS * 2;
  cv[REFI*2+0] = gx; cv[REFI*2+1] = gy;
  for (int j = 0; j < 155; ++j) {
    const float* d = dists + (size_t)(REFI + j) * 3;
    gx += F[6]*d[0]; gy += F[7]*d[1]; gz += F[8]*d[2];
    cv[(REFI+1+j)*2+0] = gx; cv[(REFI+1+j)*2+1] = gy;
    if (j < 154) { mm3(Rb + (size_t)(REFI + j) * 9, F, T);
      #pragma unroll
      for (int i = 0; i < 9; ++i) F[i] = T[i]; }
  }
  // backward: B0 = -(R[99]^T @ F0)
  float Bm[9];
  mm3T(Rb + (size_t)(REFI - 1) * 9, F0, Bm);
  #pragma unroll
  for (int i = 0; i < 9; ++i) Bm[i] = -Bm[i];
  gx = x0; gy = y0; gz = z0;
  for (int j = 0; j < 100; ++j) {
    const float* d = dists + (size_t)(99 - j) * 3;
    gx += Bm[6]*d[0]; gy += Bm[7]*d[1]; gz += Bm[8]*d[2];
    cv[(99-j)*2+0] = gx; cv[(99-j)*2+1] = gy;
    if (j < 99) { mm3T(Rb + (size_t)(98 - j) * 9, Bm, T);
      #pragma unroll
      for (int i = 0; i < 9; ++i) Bm[i] = T[i]; }
  }
}

// ---------------------------------------------------------------- C: centered DFT matrices
__global__ void k_dftmat(float* __restrict__ Mre, float* __restrict__ Mim,
                         float* __restrict__ Ire, float* __restrict__ Iim) {
  int n = blockIdx.x * blockDim.x + threadIdx.x;
  if (n >= IMG) return;
  int k = n / NPX, j = n % NPX;
  int prod = (k - 96) * (j - 96);
  int m = prod % NPX; if (m < 0) m += NPX;         // exact phase reduction mod N
  float t = (6.283185307179586f / (float)NPX) * (float)m;
  float c = cosf(t), s = sinf(t);
  Mre[n] = c;            Mim[n] = -s;              // e^{-i t}
  Ire[n] = c / (float)NPX; Iim[n] = s / (float)NPX; // e^{+i t} / N  (per-axis 1/N)
}

// ---------------------------------------------------------------- D: gaussian splat einsum (WMMA f32)
__global__ void __launch_bounds__(128)
k_splat(const float* __restrict__ curve, float* __restrict__ imgs) {
  __shared__ float gxs[32][33];
  __shared__ float gys[32][33];
  __shared__ float cxs[32], cys[32];
  const int t = threadIdx.x;
  const int b = blockIdx.z;
  const int xBase = blockIdx.x * 32, yBase = blockIdx.y * 32;
  const int lane = t & 31, wv = t >> 5;
  const int mOff = (wv >> 1) * 16, nOff = (wv & 1) * 16;
  const int hk = (lane >> 4) * 2, ml = lane & 15;

  const float inv2s2 = 200.0f;               // 1/(2*0.05^2)
  const float step = 2.0f / 191.0f;
  const int xi = t & 31, pr = t >> 5;
  const float xv = -1.f + step * (float)(xBase + xi);
  const float yv = -1.f + step * (float)(yBase + xi);
  const float* cvb = curve + (size_t)b * NPTS * 2;

  v8f acc = {0.f,0.f,0.f,0.f,0.f,0.f,0.f,0.f};
  for (int p0 = 0; p0 < NPTS; p0 += 32) {
    __syncthreads();
    if (t < 32) { cxs[t] = cvb[(p0 + t) * 2 + 0]; cys[t] = cvb[(p0 + t) * 2 + 1]; }
    __syncthreads();
    #pragma unroll
    for (int it = 0; it < 8; ++it) {
      int p = pr + it * 4;
      float dx = xv - cxs[p]; gxs[p][xi] = __expf(-dx * dx * inv2s2);
      float dy = yv - cys[p]; gys[p][xi] = __expf(-dy * dy * inv2s2);
    }
    __syncthreads();
    #pragma unroll
    for (int kk = 0; kk < 8; ++kk) {
      int ks = kk * 4 + hk;
      v2f a  = { gxs[ks][mOff + ml], gxs[ks + 1][mOff + ml] };
      v2f bb = { gys[ks][nOff + ml], gys[ks + 1][nOff + ml] };
      acc = wmma4(a, bb, acc);
    }
  }
  float* C = imgs + (size_t)b * IMG;
  #pragma unroll
  for (int i = 0; i < 8; ++i) {
    int m = mOff + i + ((lane >> 4) << 3);
    int nn = nOff + ml;
    C[(xBase + m) * NPX + (yBase + nn)] = acc[i];
  }
}

// ---------------------------------------------------------------- E: complex GEMM (WMMA f32), 192x192x192, batched
// 64x64 block tile, 4 waves, 32x32 per wave (2x2 fragments) -> 1 DS op per 2 WMMAs.
// mode 0: store re/im   mode 1: *= CTF(b,row,col) then store   mode 2: store |.| to outMag
__global__ void __launch_bounds__(128)
k_cgemm(const float* __restrict__ Are, const float* __restrict__ Aim, long long aStride,
        const float* __restrict__ Bre, const float* __restrict__ Bim, long long bStride,
        float* __restrict__ Cre, float* __restrict__ Cim, long long cStride,
        int mode, const float* __restrict__ inp, float* __restrict__ outMag) {
  __shared__ float Asr[32][65];
  __shared__ float Asi[32][65];
  __shared__ float Bsr[32][65];
  __shared__ float Bsi[32][65];
  const int t = threadIdx.x;
  const int z = blockIdx.z;
  const int mBase = blockIdx.x * 64, nBase = blockIdx.y * 64;
  const bool hasAi = (Aim != nullptr), hasBi = (Bim != nullptr);

  const float* Ar = Are + (size_t)z * aStride;
  const float* Ai = hasAi ? (Aim + (size_t)z * aStride) : nullptr;
  const float* Br = Bre + (size_t)z * bStride;
  const float* Bi = hasBi ? (Bim + (size_t)z * bStride) : nullptr;

  const int lane = t & 31, wv = t >> 5;
  const int mW = (wv >> 1) * 32, nW = (wv & 1) * 32;
  const int hk = (lane >> 4) * 2, ml = lane & 15;
  const int tc = t & 31, tg = t >> 5;          // A-tile loader coords
  const int tn = t & 63, tr = t >> 6;          // B-tile loader coords

  v8f aR[4], aI[4];
  #pragma unroll
  for (int s = 0; s < 4; ++s) {
    aR[s] = (v8f){0.f,0.f,0.f,0.f,0.f,0.f,0.f,0.f};
    aI[s] = (v8f){0.f,0.f,0.f,0.f,0.f,0.f,0.f,0.f};
  }

  for (int k0 = 0; k0 < NPX; k0 += 32) {
    __syncthreads();
    if (k0 + 32 < NPX) {   // prefetch next chunk (global_prefetch_b8)
      __builtin_prefetch(&Ar[(size_t)(mBase + tg) * NPX + k0 + 32 + tc], 0, 3);
      __builtin_prefetch(&Br[(size_t)(k0 + 32 + tr) * NPX + nBase + tn], 0, 3);
    }
    #pragma unroll
    for (int it = 0; it < 16; ++it) {
      int m = tg + it * 4;                                   // A tile -> LDS as [k][m]
      Asr[tc][m] = Ar[(size_t)(mBase + m) * NPX + k0 + tc];
      if (hasAi) Asi[tc][m] = Ai[(size_t)(mBase + m) * NPX + k0 + tc];
      int k = tr + it * 2;                                   // B tile -> LDS as [k][n]
      Bsr[k][tn] = Br[(size_t)(k0 + k) * NPX + nBase + tn];
      if (hasBi) Bsi[k][tn] = Bi[(size_t)(k0 + k) * NPX + nBase + tn];
    }
    __syncthreads();
    #pragma unroll
    for (int kk = 0; kk < 8; ++kk) {
      int ks = kk * 4 + hk;
      v2f arf[2], aif[2], brf[2], bif[2];
      #pragma unroll
      for (int mi = 0; mi < 2; ++mi) {
        arf[mi] = (v2f){ Asr[ks][mW + mi*16 + ml], Asr[ks + 1][mW + mi*16 + ml] };
        if (hasAi) aif[mi] = (v2f){ Asi[ks][mW + mi*16 + ml], Asi[ks + 1][mW + mi*16 + ml] };
      }
      #pragma unroll
      for (int ni = 0; ni < 2; ++ni) {
        brf[ni] = (v2f){ Bsr[ks][nW + ni*16 + ml], Bsr[ks + 1][nW + ni*16 + ml] };
        if (hasBi) bif[ni] = (v2f){ Bsi[ks][nW + ni*16 + ml], Bsi[ks + 1][nW + ni*16 + ml] };
      }
      #pragma unroll
      for (int mi = 0; mi < 2; ++mi)
        #pragma unroll
        for (int ni = 0; ni < 2; ++ni) {
          int s = mi * 2 + ni;
          aR[s] = wmma4(arf[mi], brf[ni], aR[s]);
          if (hasBi) {
            aI[s] = wmma4(arf[mi], bif[ni], aI[s]);
            if (hasAi) {
              v2f nai = (v2f){ -aif[mi].x, -aif[mi].y };
              aR[s] = wmma4(nai, bif[ni], aR[s]);   // re -= Ai*Bi (F32 WMMA lacks A/B NEG)
            }
          }
          if (hasAi) aI[s] = wmma4(aif[mi], brf[ni], aI[s]);
        }
    }
  }

  float* Cr = Cre + (size_t)z * cStride;
  float* Ci = Cim + (size_t)z * cStride;
  const float lam = 12.2643247f / sqrtf(300000.0f * (1.0f + 300000.0f * 9.78466e-07f));
  const float cs  = 2.7e7f;
  float Df = (mode == 1) ? inp[(size_t)z * INROW + NEIG + 6] : 0.f;

  #pragma unroll
  for (int mi = 0; mi < 2; ++mi)
    #pragma unroll
    for (int ni = 0; ni < 2; ++ni) {
      int s = mi * 2 + ni;
      #pragma unroll
      for (int i = 0; i < 8; ++i) {
        int row = mBase + mW + mi*16 + i + ((lane >> 4) << 3);
        int col = nBase + nW + ni*16 + ml;
        float r = aR[s][i], im = aI[s][i];
        if (mode == 1) {
          float fi = 0.5f * (float)(row - 96);
          float fj = 0.5f * (float)(col - 96);
          float s2 = fi * fi + fj * fj;
          float g = 6.283185307179586f * (-0.5f * Df * lam * s2
                                          + 0.25f * cs * lam * lam * lam * s2 * s2);
          float ctf = -(0.99498743710662f * sinf(g) + 0.1f * cosf(g));
          r *= ctf; im *= ctf;
        }
        if (mode == 2) {
          outMag[(size_t)z * IMG + (size_t)row * NPX + col] = sqrtf(r * r + im * im);
        } else {
          Cr[(size_t)row * NPX + col] = r;
          Ci[(size_t)row * NPX + col] = im;
        }
      }
    }
}

// ================================================================ launch
extern "C" void kernel_launch(void* const* d_in, const int* in_sizes, int n_in,
                              void* d_out, int out_size, void* d_ws, size_t ws_size,
                              hipStream_t stream) {
  const float* inp   = (const float*)d_in[0];
  const float* Gpsi  = (const float*)d_in[1];
  const float* Gth   = (const float*)d_in[2];
  const float* Psi   = (const float*)d_in[3];
  const float* Theta = (const float*)d_in[4];
  const float* dists = (const float*)d_in[5];
  const int*   pidx  = (const int*)  d_in[6];
  float* out = (float*)d_out;

  float* w = (float*)d_ws;
  size_t off = 0;
  float* psiB  = w + off; off += (size_t)BATCH * NANG;
  float* thB   = w + off; off += (size_t)BATCH * NANG;
  float* Rbuf  = w + off; off += (size_t)BATCH * NANG * 9;
  float* curve = w + off; off += (size_t)BATCH * NPTS * 2;
  float* Mre   = w + off; off += IMG;
  float* Mim   = w + off; off += IMG;
  float* Ire   = w + off; off += IMG;
  float* Iim   = w + off; off += IMG;
  float* imgs  = w + off; off += (size_t)BATCH * IMG;
  float* Pre   = w + off; off += (size_t)BATCH * IMG;
  float* Pim   = w + off; off += (size_t)BATCH * IMG;
  float* Qre   = w + off; off += (size_t)BATCH * IMG;
  float* Qim   = w + off; off += (size_t)BATCH * IMG;

  (void)in_sizes; (void)n_in; (void)out_size; (void)ws_size;

  // A1/A2: angles and rotation matrices
  {
    int n = BATCH * NANG;
    k_angles<<<dim3((n + 127) / 128), dim3(128), 0, stream>>>(inp, Gpsi, Gth, Psi, Theta, pidx, psiB, thB);
    k_rots  <<<dim3((n + 127) / 128), dim3(128), 0, stream>>>(psiB, thB, Rbuf);
  }
  // B: chain scans -> curve (256 pts per batch element)
  k_scan<<<dim3(2), dim3(128), 0, stream>>>(inp, Rbuf, dists, curve);
  // C: centered DFT matrices
  k_dftmat<<<dim3((IMG + 127) / 128), dim3(128), 0, stream>>>(Mre, Mim, Ire, Iim);
  // D: gaussian splat -> imgs  (WMMA f32 16x16x4, K = 256 curve points)
  k_splat<<<dim3(6, 6, BATCH), dim3(128), 0, stream>>>(curve, imgs);
  // E1: P = M @ imgs        (complex x real)
  k_cgemm<<<dim3(3, 3, BATCH), dim3(128), 0, stream>>>(
      Mre, Mim, 0LL, imgs, nullptr, (long long)IMG, Pre, Pim, (long long)IMG,
      0, nullptr, nullptr);
  // E2: Q = ctf .* (P @ M)  == filt * fft2_center(imgs)   (fused CTF epilogue)
  k_cgemm<<<dim3(3, 3, BATCH), dim3(128), 0, stream>>>(
      Pre, Pim, (long long)IMG, Mre, Mim, 0LL, Qre, Qim, (long long)IMG,
      1, inp, nullptr);
  // E3: P = Minv @ Q
  k_cgemm<<<dim3(3, 3, BATCH), dim3(128), 0, stream>>>(
      Ire, Iim, 0LL, Qre, Qim, (long long)IMG, Pre, Pim, (long long)IMG,
      0, nullptr, nullptr);
  // E4: out = |P @ Minv|    == abs(ifft2_center(...))     (fused magnitude epilogue)
  k_cgemm<<<dim3(3, 3, BATCH), dim3(128), 0, stream>>>(
      Pre, Pim, (long long)IMG, Ire, Iim, 0LL, Qre, Qim, (long long)IMG,
      2, nullptr, out);
}